// DecoderRNN_9560597201328
// MI455X (gfx1250) — compile-verified
//
#include <hip/hip_runtime.h>
#include <hip/hip_bf16.h>
#include <cmath>

// ---------------------------------------------------------------------------
// Greedy autoregressive LSTM decoder for MI455X (gfx1250).
//  V=32000, E=H=1024, B=16, T=128.  All math kept in f32 (argmax-driven decode)
//  using V_WMMA_F32_16X16X4_F32: M=16 == batch, N=16 tile, K=4 per instruction.
//  All weights (163 MB f32) fit in the 192 MB global L2 -> after step 0 the
//  sequential chain is L2-resident; HBM traffic is only the 262 MB output.
// ---------------------------------------------------------------------------

#define Bdim   16
#define Edim   1024
#define Hdim   1024
#define Vdim   32000
#define Tsteps 128
#define SOS_TOK 1

typedef float v2f __attribute__((ext_vector_type(2)));
typedef float v8f __attribute__((ext_vector_type(8)));

__device__ __forceinline__ v8f wmma_f32(v2f a, v2f b, v8f c) {
    // (neg_a, A, neg_b, B, c_mod, C, reuse_a, reuse_b)
    return __builtin_amdgcn_wmma_f32_16x16x4_f32(false, a, false, b,
                                                 (short)0, c, false, false);
}

__device__ __forceinline__ float sigmoidf_(float x) {
    return 1.0f / (1.0f + __expf(-x));
}

// ---------------------------------------------------------------------------
// Kernel 1: gates GEMM + LSTM cell update.
// One workgroup (128 thr = 4 waves) per 16-column slice of H.
// Wave w computes gate w (i,f,g,o) tile [16 x 16] with K = E + H = 2048,
// folding the embedding gather into the A-fragment load. Pointwise update via LDS.
// ---------------------------------------------------------------------------
__global__ __launch_bounds__(128) void gates_lstm_kernel(
    const float* __restrict__ emb,     // [V,E]
    const float* __restrict__ W_ih,    // [4H,E] row-major
    const float* __restrict__ W_hh,    // [4H,H] row-major
    const float* __restrict__ b_ih,    // [4H]
    const float* __restrict__ b_hh,    // [4H]
    const int*   __restrict__ tok,     // [B]
    const float* __restrict__ h_prev,  // [B,H]
    const float* __restrict__ c_prev,  // [B,H]
    float* __restrict__ h_next,        // [B,H]
    float* __restrict__ c_next)        // [B,H]
{
    __shared__ float lds_g[4][16][16];

    const int tid  = threadIdx.x;
    const int wave = tid >> 5;          // 0..3 -> gate index (i,f,g,o)
    const int lane = tid & 31;
    const int half = lane >> 4;         // K sub-offset selector for f32 WMMA
    const int ln   = lane & 15;         // M (A) / N (B,C) index
    const int n0   = blockIdx.x * 16;   // column tile within H
    const int kh   = 2 * half;          // A/B fragment K offset for this lane half

    const int wrow = wave * Hdim + n0 + ln;               // row in 4H dim
    const float* xrow   = emb  + (size_t)tok[ln] * Edim;  // gathered embedding row (b = ln)
    const float* hrow   = h_prev + (size_t)ln * Hdim;
    const float* wihrow = W_ih + (size_t)wrow * Edim;
    const float* whhrow = W_hh + (size_t)wrow * Hdim;

    v8f acc0 = {};
    v8f acc1 = {};
    // K over E (embedded input), 2 accumulators to break the WMMA RAW chain
    for (int k = 0; k < Edim; k += 8) {
        v2f a0 = *(const v2f*)(xrow   + k + kh);
        v2f b0 = *(const v2f*)(wihrow + k + kh);
        acc0 = wmma_f32(a0, b0, acc0);
        v2f a1 = *(const v2f*)(xrow   + k + 4 + kh);
        v2f b1 = *(const v2f*)(wihrow + k + 4 + kh);
        acc1 = wmma_f32(a1, b1, acc1);
    }
    // K over H (recurrent)
    for (int k = 0; k < Hdim; k += 8) {
        v2f a0 = *(const v2f*)(hrow   + k + kh);
        v2f b0 = *(const v2f*)(whhrow + k + kh);
        acc0 = wmma_f32(a0, b0, acc0);
        v2f a1 = *(const v2f*)(hrow   + k + 4 + kh);
        v2f b1 = *(const v2f*)(whhrow + k + 4 + kh);
        acc1 = wmma_f32(a1, b1, acc1);
    }
    v8f acc = acc0 + acc1;

    const float bias = b_ih[wrow] + b_hh[wrow];
    // C/D layout: element (m = r + 8*half, n = ln) in VGPR r
    #pragma unroll
    for (int r = 0; r < 8; ++r)
        lds_g[wave][r + 8 * half][ln] = acc[r] + bias;

    __syncthreads();

    // LSTM pointwise: 256 (b, j) pairs, 2 per thread
    for (int p = tid; p < 256; p += 128) {
        const int b = p >> 4;
        const int j = p & 15;
        const int col = n0 + j;
        const float iv = lds_g[0][b][j];
        const float fv = lds_g[1][b][j];
        const float gv = lds_g[2][b][j];
        const float ov = lds_g[3][b][j];
        const float cold = c_prev[b * Hdim + col];
        const float cn = sigmoidf_(fv) * cold + sigmoidf_(iv) * tanhf(gv);
        const float hn = sigmoidf_(ov) * tanhf(cn);
        c_next[b * Hdim + col] = cn;
        h_next[b * Hdim + col] = hn;
    }
}

// ---------------------------------------------------------------------------
// Kernel 2: logits = h @ W_lin^T + b_lin   -> [16, 32000]
// 2000 waves (250 WGs x 8 waves), each owns a 16x16 tile of V; K = H = 1024.
// ---------------------------------------------------------------------------
__global__ __launch_bounds__(256) void logits_kernel(
    const float* __restrict__ h,       // [B,H]
    const float* __restrict__ W_lin,   // [V,H] row-major
    const float* __restrict__ b_lin,   // [V]
    float* __restrict__ logits)        // [B,V]
{
    const int tid  = threadIdx.x;
    const int lane = tid & 31;
    const int half = lane >> 4;
    const int ln   = lane & 15;
    const int w    = blockIdx.x * 8 + (tid >> 5);  // 0..1999, exact (no tail)
    const int n0   = w * 16;
    const int kh   = 2 * half;

    const float* hrow = h + (size_t)ln * Hdim;
    const float* wrow = W_lin + (size_t)(n0 + ln) * Hdim;

    v8f acc0 = {};
    v8f acc1 = {};
    for (int k = 0; k < Hdim; k += 8) {
        v2f a0 = *(const v2f*)(hrow + k + kh);
        v2f b0 = *(const v2f*)(wrow + k + kh);
        acc0 = wmma_f32(a0, b0, acc0);
        v2f a1 = *(const v2f*)(hrow + k + 4 + kh);
        v2f b1 = *(const v2f*)(wrow + k + 4 + kh);
        acc1 = wmma_f32(a1, b1, acc1);
    }
    v8f acc = acc0 + acc1;

    const float bias = b_lin[n0 + ln];
    #pragma unroll
    for (int r = 0; r < 8; ++r)
        logits[(size_t)(r + 8 * half) * Vdim + n0 + ln] = acc[r] + bias;
}

// ---------------------------------------------------------------------------
// Kernel 3: per-row online max / first-argmax / sum-exp over V=32000.
// One workgroup per batch row. Writes tok (feeds next step), max, log(sum).
// ---------------------------------------------------------------------------
__global__ __launch_bounds__(256) void stats_kernel(
    const float* __restrict__ logits,  // [B,V]
    int*   __restrict__ tok,           // [B]
    float* __restrict__ rmax,          // [B]
    float* __restrict__ rlogs)         // [B]
{
    __shared__ float sm[256];
    __shared__ float ss[256];
    __shared__ int   sa[256];

    const int b   = blockIdx.x;
    const int tid = threadIdx.x;

    float m = -INFINITY;
    float s = 0.0f;
    int   am = 0;
    for (int v = tid; v < Vdim; v += 256) {
        const float x = logits[(size_t)b * Vdim + v];
        if (x > m) {                      // strict > keeps first occurrence
            s = s * __expf(m - x) + 1.0f; // exp(-inf)=0 handles the init case
            m = x;
            am = v;
        } else {
            s += __expf(x - m);
        }
    }
    sm[tid] = m; ss[tid] = s; sa[tid] = am;
    __syncthreads();

    for (int off = 128; off > 0; off >>= 1) {
        if (tid < off) {
            const float m1 = sm[tid],       s1 = ss[tid];       const int a1 = sa[tid];
            const float m2 = sm[tid + off], s2 = ss[tid + off]; const int a2 = sa[tid + off];
            const float M = fmaxf(m1, m2);
            const float S = s1 * __expf(m1 - M) + s2 * __expf(m2 - M);
            int A;
            if      (m1 > m2) A = a1;
            else if (m2 > m1) A = a2;
            else              A = (a1 < a2) ? a1 : a2;   // jnp.argmax: first index
            sm[tid] = M; ss[tid] = S; sa[tid] = A;
        }
        __syncthreads();
    }
    if (tid == 0) {
        tok[b]   = sa[0];
        rmax[b]  = sm[0];
        rlogs[b] = logf(ss[0]);
    }
}

// ---------------------------------------------------------------------------
// Kernel 4: log_softmax write for step t:  out[b,t,v] = x - max - log(sum)
// ---------------------------------------------------------------------------
__global__ __launch_bounds__(256) void writelogp_kernel(
    const float* __restrict__ logits,  // [B,V]
    const float* __restrict__ rmax,
    const float* __restrict__ rlogs,
    float* __restrict__ out,           // [B,T,V]
    int t)
{
    const int idx = blockIdx.x * 256 + threadIdx.x;  // < B*V (exact grid)
    const int b = idx / Vdim;
    const int v = idx - b * Vdim;
    out[((size_t)b * Tsteps + t) * Vdim + v] =
        logits[(size_t)b * Vdim + v] - rmax[b] - rlogs[b];
}

// ---------------------------------------------------------------------------
// Init / final copy
// ---------------------------------------------------------------------------
__global__ __launch_bounds__(256) void init_kernel(
    const float* __restrict__ h0, const float* __restrict__ c0,
    float* __restrict__ h, float* __restrict__ c, int* __restrict__ tok)
{
    const int idx = blockIdx.x * 256 + threadIdx.x;
    if (idx < Bdim * Hdim) { h[idx] = h0[idx]; c[idx] = c0[idx]; }
    if (idx < Bdim) tok[idx] = SOS_TOK;
}

__global__ __launch_bounds__(256) void final_copy_kernel(
    const float* __restrict__ h, const float* __restrict__ c,
    float* __restrict__ out)
{
    const int idx = blockIdx.x * 256 + threadIdx.x;
    const size_t base = (size_t)Bdim * Tsteps * Vdim;
    if (idx < Bdim * Hdim) {
        out[base + idx]               = h[idx];
        out[base + Bdim * Hdim + idx] = c[idx];
    }
}

// ---------------------------------------------------------------------------
extern "C" void kernel_launch(void* const* d_in, const int* in_sizes, int n_in,
                              void* d_out, int out_size, void* d_ws, size_t ws_size,
                              hipStream_t stream) {
    // d_in order (setup_inputs): 0 encoder_outputs (unused beyond shape),
    // 1 h0 [1,B,H], 2 c0 [1,B,H], 3 embedding [V,E], 4 W_ih [4H,E],
    // 5 W_hh [4H,H], 6 b_ih [4H], 7 b_hh [4H], 8 W_lin [V,H], 9 b_lin [V]
    const float* h0    = (const float*)d_in[1];
    const float* c0    = (const float*)d_in[2];
    const float* emb   = (const float*)d_in[3];
    const float* W_ih  = (const float*)d_in[4];
    const float* W_hh  = (const float*)d_in[5];
    const float* b_ih  = (const float*)d_in[6];
    const float* b_hh  = (const float*)d_in[7];
    const float* W_lin = (const float*)d_in[8];
    const float* b_lin = (const float*)d_in[9];
    float* out = (float*)d_out;

    // Workspace layout (floats): ~2.3 MB total
    float* ws = (float*)d_ws;
    float* hb0 = ws;                         // [B*H]
    float* hb1 = ws + 16384;
    float* cb0 = ws + 32768;
    float* cb1 = ws + 49152;
    float* logits = ws + 65536;              // [B*V] = 512000
    float* rmax   = ws + 65536 + 512000;     // [B]
    float* rlogs  = rmax + Bdim;             // [B]
    int*   tok    = (int*)(rlogs + Bdim);    // [B]
    float* hb[2] = { hb0, hb1 };
    float* cb[2] = { cb0, cb1 };

    init_kernel<<<64, 256, 0, stream>>>(h0, c0, hb[0], cb[0], tok);

    for (int t = 0; t < Tsteps; ++t) {
        const int p = t & 1;
        gates_lstm_kernel<<<Hdim / 16, 128, 0, stream>>>(
            emb, W_ih, W_hh, b_ih, b_hh, tok,
            hb[p], cb[p], hb[1 - p], cb[1 - p]);
        logits_kernel<<<(Vdim / 16) / 8, 256, 0, stream>>>(
            hb[1 - p], W_lin, b_lin, logits);
        stats_kernel<<<Bdim, 256, 0, stream>>>(logits, tok, rmax, rlogs);
        writelogp_kernel<<<(Bdim * Vdim) / 256, 256, 0, stream>>>(
            logits, rmax, rlogs, out, t);
    }
    // Final h/c: after 128 steps the live buffers are index 0 again.
    final_copy_kernel<<<64, 256, 0, stream>>>(hb[0], cb[0], out);
}